// SetConv_69028714381387
// MI455X (gfx1250) — compile-verified
//
#include <hip/hip_runtime.h>

// ---------------------------------------------------------------------------
// MI455X / gfx1250 fused GNN "SetConv" pipeline, fp32 WMMA (V_WMMA_F32_16X16X4_F32)
//
//   S   = segment_sum(x, batch)                      [10000,128]
//   Z0  = S @ W_lin            (+ col stats)         [10000,128]
//   bn0 -> per-col scale a0 / shift c0
//   Z1  = (x + relu(bn0(Z0[batch]))) @ W1 + b1       [500000,128]  (stored in d_out)
//   bn1 -> a1/c1
//   out = relu(bn1(Z1)) @ W2 + b2                    [500000,128]  (in-place on d_out)
// ---------------------------------------------------------------------------

#define NF   128        // feature dim (NIN == NOUT == 128)
#define LDA  132        // padded LDS row stride in floats (bank-conflict avoidance)
#define EPSV 1e-5f

typedef __attribute__((ext_vector_type(2))) float v2f;
typedef __attribute__((ext_vector_type(8))) float v8f;

__global__ void zero_f32(float* p, int n) {
    int i = blockIdx.x * blockDim.x + threadIdx.x;
    if (i < n) p[i] = 0.f;
}

// Segment-sum pooling. batch is sorted, so accumulate in registers and flush
// with an atomic only when the graph id changes (≈ boundaries-per-strip atomics
// instead of one per node). 128 threads = one feature column each (coalesced
// 512B row reads), each block walks a strip of consecutive nodes.
__global__ void segsum_kernel(const float* __restrict__ x, const int* __restrict__ batch,
                              float* __restrict__ S, int M, int rowsPerBlock) {
    const int col = threadIdx.x;                     // 0..127
    int r0 = blockIdx.x * rowsPerBlock;
    int r1 = r0 + rowsPerBlock; if (r1 > M) r1 = M;
    float acc = 0.f;
    int   cur = -1;
    for (int r = r0; r < r1; ++r) {
        const int g = batch[r];                      // scalar (uniform) load
        const float v = x[(long long)r * NF + col];
        if (g != cur) {
            if (cur >= 0) atomicAdd(&S[(long long)cur * NF + col], acc);
            acc = 0.f; cur = g;
        }
        acc += v;
    }
    if (cur >= 0) atomicAdd(&S[(long long)cur * NF + col], acc);
}

// Turn column sums / sums-of-squares into fused BN scale/shift:
//   a = g * rsqrt(var + eps),  c = b - mu * a   (biased variance, as reference)
__global__ void bn_finalize_kernel(const float* __restrict__ sum, const float* __restrict__ sumsq,
                                   const float* __restrict__ g,   const float* __restrict__ b,
                                   float* __restrict__ a, float* __restrict__ c, float invM) {
    const int n = threadIdx.x;                       // 128 threads
    const float mu  = sum[n]  * invM;
    const float var = sumsq[n] * invM - mu * mu;
    const float r   = rsqrtf(var + EPSV);
    const float an  = g[n] * r;
    a[n] = an;
    c[n] = b[n] - mu * an;
}

// ---------------------------------------------------------------------------
// Fused 128-wide GEMM:  Out[m, :] = prologue(m, :) @ W + bias   (K = N = 128)
//
// Block = 256 threads = 8 wave32s. Wave w owns output columns [16w, 16w+16).
// B fragments (its 16 columns of W, all 128 K) are preloaded once into
// 32 x v2f = 64 VGPRs and live there for the whole kernel.
// Per 16-row M-tile: the block stages the fused A tile (16x128) into LDS
// (row stride 132 floats -> ds_load_b64 fragment reads are conflict-free),
// then each wave runs 32 x V_WMMA_F32_16X16X4_F32 into an 8-VGPR accumulator.
//
// Assumed f32 WMMA operand layouts (ISA 7.12.2):
//   A 16x4 : lanes 0-15 rows M=lane, .x=K0 .y=K1 ; lanes 16-31 .x=K2 .y=K3
//   B 4x16 : mirrored (lanes 0-15 cols, .x=K0 .y=K1 ; lanes 16-31 .x=K2 .y=K3)
//   C/D    : vgpr r -> row r (lanes 0-15) / r+8 (lanes 16-31), col = lane&15
//
// PRE = 0 : A = Ax                                       (plain GEMM)
// PRE = 1 : A = Ax + relu(Zg[batch[m]] * bnA + bnC)      (gather + BN0 + ReLU + add)
// PRE = 2 : A = relu(Ax * bnA + bnC)                     (BN1 + ReLU; in-place safe)
// STATS   : accumulate per-column sum / sum-of-squares of Out (pre-BN stats)
// ---------------------------------------------------------------------------
template<int PRE, bool STATS>
__global__ void __launch_bounds__(256)
gemm128_kernel(const float* Ax, const float* __restrict__ Zg, const int* __restrict__ batch,
               const float* __restrict__ bnA, const float* __restrict__ bnC,
               const float* __restrict__ W,   const float* __restrict__ bias,
               float* Out, float* __restrict__ colsum, float* __restrict__ colsumsq, int M) {
    __shared__ float ldsA[16 * LDA];
    __shared__ float ldsBnA[NF];
    __shared__ float ldsBnC[NF];

    const int tid  = threadIdx.x;
    const int lane = tid & 31;
    const int wave = tid >> 5;                 // 0..7
    const int n0   = wave * 16;
    const int n    = n0 + (lane & 15);         // this lane's output column

    if (PRE != 0) {
        if (tid < NF) { ldsBnA[tid] = bnA[tid]; ldsBnC[tid] = bnC[tid]; }
    }

    // ---- preload B fragments (held in VGPRs for the whole kernel) ----
    v2f Bfrag[32];
    {
        const int koff = (lane >> 4) * 2;      // lanes 16-31 hold K+2/K+3
        #pragma unroll
        for (int kk = 0; kk < 32; ++kk) {
            const int k = kk * 4 + koff;
            v2f b;
            b.x = W[(k + 0) * NF + n];
            b.y = W[(k + 1) * NF + n];
            Bfrag[kk] = b;
        }
    }
    const float bval = bias ? bias[n] : 0.f;

    float s1 = 0.f, s2 = 0.f;                  // per-lane column stats partials

    const int numTiles = M >> 4;
    for (int t = blockIdx.x; t < numTiles; t += gridDim.x) {
        const long long m0 = (long long)t * 16;

        __syncthreads();                       // LDS A free to overwrite (also orders ldsBn*)

        // ---- stage fused A tile to LDS: 2048 floats, 8 per thread ----
        {
            const int row = tid >> 4;          // 0..15
            const int k0  = (tid & 15) * 8;    // 0,8,...,120
            const long long node = m0 + row;
            const float4* xp = (const float4*)(Ax + node * NF + k0);
            const float4 xa = xp[0], xb = xp[1];
            float xv[8] = {xa.x, xa.y, xa.z, xa.w, xb.x, xb.y, xb.z, xb.w};
            float ov[8];
            if (PRE == 1) {
                const long long g = (long long)batch[node];
                const float4* zp = (const float4*)(Zg + g * NF + k0);
                const float4 za = zp[0], zb = zp[1];
                float zv[8] = {za.x, za.y, za.z, za.w, zb.x, zb.y, zb.z, zb.w};
                #pragma unroll
                for (int j = 0; j < 8; ++j) {
                    float tv = zv[j] * ldsBnA[k0 + j] + ldsBnC[k0 + j];
                    ov[j] = xv[j] + (tv > 0.f ? tv : 0.f);
                }
            } else if (PRE == 2) {
                #pragma unroll
                for (int j = 0; j < 8; ++j) {
                    float tv = xv[j] * ldsBnA[k0 + j] + ldsBnC[k0 + j];
                    ov[j] = tv > 0.f ? tv : 0.f;
                }
            } else {
                #pragma unroll
                for (int j = 0; j < 8; ++j) ov[j] = xv[j];
            }
            #pragma unroll
            for (int j = 0; j < 8; ++j) ldsA[row * LDA + k0 + j] = ov[j];
        }
        __syncthreads();

        // ---- 16x16x128 via 32 chained f32 WMMAs ----
        v8f acc = {0.f, 0.f, 0.f, 0.f, 0.f, 0.f, 0.f, 0.f};
        {
            const int m    = lane & 15;
            const int koff = (lane >> 4) * 2;
            const float* arow = &ldsA[m * LDA + koff];   // 8-byte aligned
            #pragma unroll
            for (int kk = 0; kk < 32; ++kk) {
                const v2f a = *(const v2f*)(arow + kk * 4);   // ds_load_b64
                acc = __builtin_amdgcn_wmma_f32_16x16x4_f32(
                        /*neg_a=*/false, a, /*neg_b=*/false, Bfrag[kk],
                        /*c_mod=*/(short)0, acc, /*reuse_a=*/false, /*reuse_b=*/false);
            }
        }

        // ---- epilogue: bias, store, per-column stats ----
        {
            const long long mbase = m0 + (long long)((lane >> 4) * 8);
            #pragma unroll
            for (int r = 0; r < 8; ++r) {
                const float v = acc[r] + bval;
                Out[(mbase + r) * NF + n] = v;
                if (STATS) { s1 += v; s2 += v * v; }
            }
        }
    }

    if (STATS) {
        // lanes l and l^16 hold the same column -> fold, then 16 atomics per wave
        s1 += __shfl_xor(s1, 16, 32);
        s2 += __shfl_xor(s2, 16, 32);
        if (lane < 16) {
            atomicAdd(&colsum[n],   s1);
            atomicAdd(&colsumsq[n], s2);
        }
    }
}

// ---------------------------------------------------------------------------
extern "C" void kernel_launch(void* const* d_in, const int* in_sizes, int n_in,
                              void* d_out, int out_size, void* d_ws, size_t ws_size,
                              hipStream_t stream) {
    const float* x     = (const float*)d_in[0];
    // d_in[1] = edge_index, d_in[2] = edge_attr : unused by the reference
    const int*   batch = (const int*)d_in[3];
    const float* W_lin = (const float*)d_in[4];
    const float* bn_g  = (const float*)d_in[5];
    const float* bn_b  = (const float*)d_in[6];
    const float* W1    = (const float*)d_in[7];
    const float* b1    = (const float*)d_in[8];
    const float* bn1_g = (const float*)d_in[9];
    const float* bn1_b = (const float*)d_in[10];
    const float* W2    = (const float*)d_in[11];
    const float* b2    = (const float*)d_in[12];

    const int M = in_sizes[0] / NF;          // 500000 nodes
    const int G = 10000;                     // N_GRAPHS (reference constant)

    // workspace layout (floats): S | Z0 | 8x128 stats  (~10.3 MB total)
    float* S    = (float*)d_ws;
    float* Z0   = S  + (size_t)G * NF;
    float* st   = Z0 + (size_t)G * NF;
    float* sum0 = st + 0 * NF; float* sq0 = st + 1 * NF;
    float* a0   = st + 2 * NF; float* c0  = st + 3 * NF;
    float* sum1 = st + 4 * NF; float* sq1 = st + 5 * NF;
    float* a1   = st + 6 * NF; float* c1  = st + 7 * NF;
    float* Z1   = (float*)d_out;             // Z1 lives in d_out (in-place-safe pass 3)

    // zero the accumulation targets (every call: harness does not re-poison)
    zero_f32<<<(G * NF + 255) / 256, 256, 0, stream>>>(S, G * NF);
    zero_f32<<<(8 * NF + 255) / 256, 256, 0, stream>>>(st, 8 * NF);

    // 1) S = segment_sum(x, batch)
    const int rowsPerBlock = 256;
    segsum_kernel<<<(M + rowsPerBlock - 1) / rowsPerBlock, NF, 0, stream>>>(
        x, batch, S, M, rowsPerBlock);

    // 2) Z0 = S @ W_lin  (+ column stats for BN0)
    const int tiles0 = G / 16;               // 625
    gemm128_kernel<0, true><<<tiles0, 256, 0, stream>>>(
        S, nullptr, nullptr, nullptr, nullptr, W_lin, nullptr, Z0, sum0, sq0, G);
    bn_finalize_kernel<<<1, NF, 0, stream>>>(sum0, sq0, bn_g, bn_b, a0, c0, 1.f / (float)G);

    // 3) Z1 = (x + relu(bn0(Z0[batch]))) @ W1 + b1  (+ stats for BN1) -> d_out
    const int tiles1 = M / 16;               // 31250
    const int grid1  = tiles1 < 2048 ? tiles1 : 2048;
    gemm128_kernel<1, true><<<grid1, 256, 0, stream>>>(
        x, Z0, batch, a0, c0, W1, b1, Z1, sum1, sq1, M);
    bn_finalize_kernel<<<1, NF, 0, stream>>>(sum1, sq1, bn1_g, bn1_b, a1, c1, 1.f / (float)M);

    // 4) out = relu(bn1(Z1)) @ W2 + b2   (in-place: each block reads its 16 rows
    //    into LDS before overwriting exactly those rows)
    gemm128_kernel<2, false><<<grid1, 256, 0, stream>>>(
        Z1, nullptr, nullptr, a1, c1, W2, b2, Z1, nullptr, nullptr, M);
}